// NeRFLoss_71382356459603
// MI455X (gfx1250) — compile-verified
//
#include <hip/hip_runtime.h>
#include <math.h>

// ---------------- problem constants ----------------
#define N_RAYS   65536
#define S        192
#define NCLS     7
#define DIST_IDX (N_RAYS * 3)        // 196608
#define CE_IDX   (N_RAYS * 3 + 1)    // 196609
#define SKY_OFF  (N_RAYS * 3 + 2)    // 196610

// ---------------- dist kernel config ----------------
#define DB   1024                    // blocks for dist kernel
#define WPB  8                       // waves per block (256 threads)
#define RPW  (N_RAYS / (DB * WPB))   // rays per wave = 8

// low 32 bits of a flat shared-memory pointer == LDS byte offset (wave's window)
__device__ __forceinline__ unsigned lds_u32(const void* p) {
    return (unsigned)(unsigned long long)p;
}

// Issue 3 async b64 global->LDS loads covering 24 bytes for this lane.
// Per CDNA5 ISA: the offset is added to both the LDS address (vdst)
//                and the global address (vaddr; saddr=off)
__device__ __forceinline__ void async_row24(unsigned ldsdst, const float* g) {
    asm volatile(
        "global_load_async_to_lds_b64 %0, %1, off\n\t"
        "global_load_async_to_lds_b64 %0, %1, off offset:8\n\t"
        "global_load_async_to_lds_b64 %0, %1, off offset:16"
        :: "v"(ldsdst), "v"(g) : "memory");
}

__device__ __forceinline__ void wait_async_le6() {
    asm volatile("s_wait_asynccnt 0x6" ::: "memory");
}
__device__ __forceinline__ void wait_async_0() {
    asm volatile("s_wait_asynccnt 0x0" ::: "memory");
}

// wave32 exclusive scan
__device__ __forceinline__ float wave_exscan(float x, int lane) {
    float v = x;
#pragma unroll
    for (int d = 1; d < 32; d <<= 1) {
        float y = __shfl_up(v, d, 32);
        if (lane >= d) v += y;
    }
    return v - x;
}

__device__ __forceinline__ float wave_sum(float x) {
#pragma unroll
    for (int m = 16; m >= 1; m >>= 1) x += __shfl_xor(x, m, 32);
    return x;
}

// ---------------------------------------------------------------------------
// Distortion loss: one wave per ray, double-buffered async global->LDS staging
// ---------------------------------------------------------------------------
__global__ __launch_bounds__(256) void dist_kernel(const float* __restrict__ ws0,
                                                   const float* __restrict__ zv,
                                                   float* __restrict__ partials) {
    __shared__ float sh_w[WPB][2][S];
    __shared__ float sh_z[WPB][2][S];
    __shared__ float sh_red[WPB];

    const int lane = threadIdx.x & 31;
    const int wave = threadIdx.x >> 5;
    const int gw   = blockIdx.x * WPB + wave;
    const int ray0 = gw * RPW;

    const unsigned ldsw[2] = { lds_u32(&sh_w[wave][0][0]) + 24u * lane,
                               lds_u32(&sh_w[wave][1][0]) + 24u * lane };
    const unsigned ldsz[2] = { lds_u32(&sh_z[wave][0][0]) + 24u * lane,
                               lds_u32(&sh_z[wave][1][0]) + 24u * lane };

    // prime buffer 0 with ray0 (6 async ops outstanding)
    {
        const float* wp = ws0 + (size_t)ray0 * S + 6 * lane;
        const float* zp = zv  + (size_t)ray0 * S + 6 * lane;
        async_row24(ldsw[0], wp);
        async_row24(ldsz[0], zp);
    }

    float acc = 0.0f;
    for (int i = 0; i < RPW; ++i) {
        const int cur = i & 1;
        if (i + 1 < RPW) {
            const int ray = ray0 + i + 1;
            const float* wp = ws0 + (size_t)ray * S + 6 * lane;
            const float* zp = zv  + (size_t)ray * S + 6 * lane;
            async_row24(ldsw[cur ^ 1], wp);
            async_row24(ldsz[cur ^ 1], zp);
            wait_async_le6();   // in-order completion => batch i is done
        } else {
            wait_async_0();
        }

        const float* wrow = &sh_w[wave][cur][0];
        const float* zrow = &sh_z[wave][cur][0];
        const int base = lane * 6;

        // z neighbors with edge clamp: z[-1]:=z[0], z[S]:=z[S-1]
        float zl[8];
#pragma unroll
        for (int j = 0; j < 8; ++j) {
            int idx = base - 1 + j;
            idx = idx < 0 ? 0 : (idx > S - 1 ? S - 1 : idx);
            zl[j] = zrow[idx];
        }

        // per-lane pass: local exclusive prefix sums + totals
        float tw = 0.0f, twm = 0.0f, lp = 0.0f, intra = 0.0f;
#pragma unroll
        for (int k = 0; k < 6; ++k) {
            const float wk  = wrow[base + k];
            const float mk  = 0.25f * (zl[k] + 2.0f * zl[k + 1] + zl[k + 2]);
            const float ik  = 0.5f  * (zl[k + 2] - zl[k]);
            const float wmk = wk * mk;
            lp    += wmk * tw - wk * twm;     // tw/twm hold exclusive sums here
            intra += wk * wk * ik;
            tw  += wk;
            twm += wmk;
        }

        // wave-level exclusive offsets across lanes (ray-wide prefix sums)
        const float ow  = wave_exscan(tw,  lane);
        const float owm = wave_exscan(twm, lane);

        acc += lp + ow * twm - owm * tw + intra * (1.0f / 3.0f);
    }

    // reduce: lanes -> wave -> block partial (deterministic)
    acc = wave_sum(acc);
    if (lane == 0) sh_red[wave] = acc;
    __syncthreads();
    if (threadIdx.x == 0) {
        float s = 0.0f;
#pragma unroll
        for (int j = 0; j < WPB; ++j) s += sh_red[j];
        partials[blockIdx.x] = s;
    }
}

// ---------------------------------------------------------------------------
// Fused elementwise: rgb squared error, CE partials, sky term
// ---------------------------------------------------------------------------
__global__ __launch_bounds__(256) void ew_kernel(const float* __restrict__ rgb0,
                                                 const float* __restrict__ sem,
                                                 const float* __restrict__ depth,
                                                 const float* __restrict__ target,
                                                 const int* __restrict__ label,
                                                 float* __restrict__ out,
                                                 float* __restrict__ ce_partials) {
    const int r = blockIdx.x * 256 + threadIdx.x;
    float ce = 0.0f;
    if (r < N_RAYS) {
#pragma unroll
        for (int c = 0; c < 3; ++c) {
            const float d = rgb0[3 * r + c] - target[3 * r + c];
            out[3 * r + c] = d * d;
        }
        float x[NCLS];
#pragma unroll
        for (int c = 0; c < NCLS; ++c) x[c] = sem[NCLS * r + c];
        float mx = x[0];
#pragma unroll
        for (int c = 1; c < NCLS; ++c) mx = fmaxf(mx, x[c]);
        float s = 0.0f;
#pragma unroll
        for (int c = 0; c < NCLS; ++c) s += expf(x[c] - mx);
        const int lb = label[r];
        float xl = 0.0f;
#pragma unroll
        for (int c = 0; c < NCLS; ++c) xl = (lb == c) ? x[c] : xl;
        ce = (mx + logf(s)) - xl;

        out[SKY_OFF + r] = (lb == 4) ? 0.1f * expf(-depth[r]) : 0.0f;
    }
    // deterministic block reduction of CE
    const int lane = threadIdx.x & 31, wave = threadIdx.x >> 5;
    ce = wave_sum(ce);
    __shared__ float red[8];
    if (lane == 0) red[wave] = ce;
    __syncthreads();
    if (threadIdx.x == 0) {
        float s = 0.0f;
#pragma unroll
        for (int j = 0; j < 8; ++j) s += red[j];
        ce_partials[blockIdx.x] = s;
    }
}

// ---------------------------------------------------------------------------
// Finalize: reduce partials, write the two scalars
// ws layout: [0..DB-1] dist partials, [1024..1279] ce partials (256 blocks)
// ---------------------------------------------------------------------------
__global__ __launch_bounds__(256) void fin_kernel(const float* __restrict__ ws,
                                                  float* __restrict__ out) {
    const int t = threadIdx.x;
    float d = ws[t] + ws[t + 256] + ws[t + 512] + ws[t + 768];
    float c = ws[1024 + t];
    const int lane = t & 31, wave = t >> 5;
    d = wave_sum(d);
    c = wave_sum(c);
    __shared__ float rd[8], rc[8];
    if (lane == 0) { rd[wave] = d; rc[wave] = c; }
    __syncthreads();
    if (t == 0) {
        float ds = 0.0f, cs = 0.0f;
#pragma unroll
        for (int j = 0; j < 8; ++j) { ds += rd[j]; cs += rc[j]; }
        out[DIST_IDX] = 1e-4f * ds / (float)N_RAYS;
        out[CE_IDX]   = 0.04f * cs / (float)N_RAYS;
    }
}

// ---------------------------------------------------------------------------
extern "C" void kernel_launch(void* const* d_in, const int* in_sizes, int n_in,
                              void* d_out, int out_size, void* d_ws, size_t ws_size,
                              hipStream_t stream) {
    const float* rgb0   = (const float*)d_in[0];
    // d_in[1] = opacity0 (unused by reference output)
    const float* ws0    = (const float*)d_in[2];
    const float* zv     = (const float*)d_in[3];
    const float* sem    = (const float*)d_in[4];
    const float* depth  = (const float*)d_in[5];
    const float* target = (const float*)d_in[6];
    const int*   label  = (const int*)d_in[7];
    // d_in[8] = stages (unused; reference path fixed at stages=1)

    float* out = (float*)d_out;
    float* wsf = (float*)d_ws;

    dist_kernel<<<DB, 256, 0, stream>>>(ws0, zv, wsf);                       // wsf[0..1023]
    ew_kernel<<<N_RAYS / 256, 256, 0, stream>>>(rgb0, sem, depth, target,
                                                label, out, wsf + 1024);     // wsf[1024..1279]
    fin_kernel<<<1, 256, 0, stream>>>(wsf, out);
}